// MiniBeamformer_90271622627704
// MI455X (gfx1250) — compile-verified
//
#include <hip/hip_runtime.h>
#include <cmath>

typedef __attribute__((ext_vector_type(2))) float v2f;
typedef __attribute__((ext_vector_type(8))) float v8f;

#define FREQS 256
#define CH_IN 8
#define CH_INNER 32
#define FREQ_K 3
#define TIME_K 5
#define DIL 2
#define T_LEN 1000
#define B_SZ 4
#define TT 32              // output time tile
#define THALO 40           // TT + (TIME_K-1)*DIL
#define NF 3               // center freq + 2 halo
#define CEPS 1e-6f
#define CS_W_ELEMS (NF * 2 * CH_IN * CH_INNER)     // 1536
#define CS_B_OFF   CS_W_ELEMS                      // biases appended
#define CS_B_ELEMS (NF * 2 * CH_INNER)             // 192
#define TC_W_ELEMS (2 * CH_IN * TIME_K * CH_INNER) // 2560

#define WMMA_F32(a, b, c) \
    __builtin_amdgcn_wmma_f32_16x16x4_f32(false, (a), false, (b), (short)0, (c), false, false)

// branch-free tanh on the TRANS pipe: 1 - 2/(exp(2x)+1); saturates via inf/0.
__device__ __forceinline__ float fast_tanh(float v) {
    float e = __expf(2.f * v);
    return 1.f - 2.f * __builtin_amdgcn_rcpf(e + 1.f);
}

// branch-free octant-folded atan2 (minimax poly + v_rcp + cndmask selects)
__device__ __forceinline__ float fast_atan2(float y, float x) {
    float ax = __builtin_fabsf(x), ay = __builtin_fabsf(y);
    float mx = fmaxf(ax, ay), mn = fminf(ax, ay);
    float a  = mn * __builtin_amdgcn_rcpf(mx);
    a = (mx == 0.f) ? 0.f : a;                 // atan2(0,0) -> 0
    float s = a * a;
    float r = a * fmaf(s, fmaf(s, fmaf(s, fmaf(s, fmaf(s, -0.0117212f,
                  0.05265332f), -0.11643287f), 0.19354346f), -0.33262347f),
                  0.99997726f);
    r = (ay > ax) ? (1.57079637f - r) : r;     // fold |y|>|x|
    r = (x < 0.f) ? (3.14159274f - r) : r;     // fold x<0
    return copysignf(r, y);
}

__global__ __launch_bounds__(128)
void minibf_fused(const float* __restrict__ x,
                  const float* __restrict__ cs_wr, const float* __restrict__ cs_wi,
                  const float* __restrict__ cs_br, const float* __restrict__ cs_bi,
                  const float* __restrict__ fcv_wr, const float* __restrict__ fcv_wi,
                  const float* __restrict__ fcv_br, const float* __restrict__ fcv_bi,
                  const float* __restrict__ tc_wr, const float* __restrict__ tc_wi,
                  const float* __restrict__ tc_br, const float* __restrict__ tc_bi,
                  const float* __restrict__ lw_r, const float* __restrict__ lw_i,
                  const float* __restrict__ lb_r, const float* __restrict__ lb_i,
                  float* __restrict__ out)
{
    // intermediates for this (freq, time-tile) region, all resident in LDS (~63.5 KB)
    __shared__ float sm_t0[NF * THALO * 2 * CH_IN];     // cLog out      (7.5 KB)
    __shared__ float sm_t1[NF * THALO * 2 * CH_INNER];  // cs+act out    (30 KB)
    __shared__ float sm_t2[THALO * 2 * CH_INNER];       // fc+act out    (10 KB)
    __shared__ float sm_w[TC_W_ELEMS];                  // cs weights THEN tc weights (10 KB union)
    __shared__ float sm_d[4][16][16];                   // WMMA D staging (4 KB)
    __shared__ float sm_t3[TT * 2 * CH_IN];             // tc+act+exp    (2 KB)

    const int tid  = threadIdx.x;
    const int lane = tid & 31;
    const int wave = tid >> 5;
    const int fc   = blockIdx.x;          // center frequency
    const int t0   = blockIdx.y * TT;     // tile start time
    const int b    = blockIdx.z;          // batch

    const long PSTRIDE = (long)CH_IN * FREQS * T_LEN;   // real->imag plane stride

    // ---- warm L2/L0 for this block's input rows (global_prefetch_b8)
    if (tid < NF * CH_IN * 2) {
        int p  = tid / (NF * CH_IN);
        int r  = tid % (NF * CH_IN);
        int fi = r / CH_IN, c = r % CH_IN;
        int f  = fc - 1 + fi;
        if (f >= 0 && f < FREQS && t0 < T_LEN) {
            long base = (((long)(b * 2 + p) * CH_IN + c) * FREQS + f) * T_LEN + t0;
            __builtin_prefetch(&x[base], 0, 0);
        }
    }

    // ---- stage cs weights+biases (zero-padded for invalid freqs): sm_w[fi][sel][ci][co]
    for (int i = tid; i < CS_W_ELEMS + CS_B_ELEMS; i += 128) {
        float val = 0.f;
        if (i < CS_W_ELEMS) {
            int fi  = i / (2 * CH_IN * CH_INNER);
            int r   = i % (2 * CH_IN * CH_INNER);
            int sel = r / (CH_IN * CH_INNER);
            int r2  = r % (CH_IN * CH_INNER);
            int ci  = r2 / CH_INNER;
            int co  = r2 % CH_INNER;
            int f   = fc - 1 + fi;
            if (f >= 0 && f < FREQS)
                val = (sel ? cs_wi : cs_wr)[(f * CH_INNER + co) * CH_IN + ci];
            sm_w[i] = val;                      // layout == linear decomposition
        } else {
            int j   = i - CS_W_ELEMS;           // [fi][sel][co]
            int fi  = j / (2 * CH_INNER);
            int sel = (j / CH_INNER) & 1;
            int co  = j % CH_INNER;
            int f   = fc - 1 + fi;
            if (f >= 0 && f < FREQS)
                val = (sel ? cs_bi : cs_br)[f * CH_INNER + co];
            sm_w[CS_B_OFF + j] = val;
        }
    }

    // ---- Phase A: cLog over (3 freqs) x (40 times) x (8 ch)
    for (int i = tid; i < NF * THALO * CH_IN; i += 128) {
        int fi = i / (THALO * CH_IN);
        int r  = i % (THALO * CH_IN);
        int tt = r / CH_IN;
        int c  = r % CH_IN;
        int f  = fc - 1 + fi;
        int t  = t0 + tt;
        float lr = 0.f, li = 0.f;
        if (f >= 0 && f < FREQS && t < T_LEN) {
            long base = (((long)(b * 2) * CH_IN + c) * FREQS + f) * T_LEN + t;
            float re = x[base];
            float im = x[base + PSTRIDE];
            float mag = __builtin_sqrtf(re * re + im * im);
            lr = __logf(mag + CEPS);
            li = fast_atan2(im, re);
        }
        sm_t0[((fi * THALO + tt) * 2 + 0) * CH_IN + c] = lr;
        sm_t0[((fi * THALO + tt) * 2 + 1) * CH_IN + c] = li;
    }
    __syncthreads();

    // ---- Phase B: per-freq complex 1x1 (8 -> 32) as WMMA + cAct.
    // 18 tasks = (3 freqs) x (3 time m-tiles) x (2 N-halves); one wave per task keeps
    // all 4 product tiles (Xr/Xi x Wr/Wi) in registers -> in-wave complex combine.
    for (int tsk = wave; tsk < NF * 3 * 2; tsk += 4) {
        int fi  = tsk / 6;
        int rem = tsk % 6;
        int mt  = rem >> 1;
        int nh  = rem & 1;
        int tt   = mt * 16 + (lane & 15);
        int ttc  = tt < THALO ? tt : (THALO - 1);   // clamp padded rows
        int koff = (lane >> 4) << 1;
        int co   = nh * 16 + (lane & 15);
        v8f d_rr = {}, d_ri = {}, d_ir = {}, d_ii = {};
#pragma unroll
        for (int kk = 0; kk < 2; kk++) {
            int ci = kk * 4 + koff;
            const float* t0p = &sm_t0[((fi * THALO + ttc) * 2) * CH_IN];
            v2f ar, ai, br_, bi_;
            ar.x = t0p[ci];            ar.y = t0p[ci + 1];
            ai.x = t0p[CH_IN + ci];    ai.y = t0p[CH_IN + ci + 1];
            br_.x = sm_w[((fi * 2 + 0) * CH_IN + ci    ) * CH_INNER + co];
            br_.y = sm_w[((fi * 2 + 0) * CH_IN + ci + 1) * CH_INNER + co];
            bi_.x = sm_w[((fi * 2 + 1) * CH_IN + ci    ) * CH_INNER + co];
            bi_.y = sm_w[((fi * 2 + 1) * CH_IN + ci + 1) * CH_INNER + co];
            d_rr = WMMA_F32(ar, br_, d_rr);
            d_ri = WMMA_F32(ar, bi_, d_ri);
            d_ir = WMMA_F32(ai, br_, d_ir);
            d_ii = WMMA_F32(ai, bi_, d_ii);
        }
        float csbr = sm_w[CS_B_OFF + (fi * 2 + 0) * CH_INNER + co];
        float csbi = sm_w[CS_B_OFF + (fi * 2 + 1) * CH_INNER + co];
        // D-row validity is UNIFORM across the 8 acc elements of a lane half:
        // row(lane, v) = rowbase + v, rowbase in {0,8,16,24,32,40}; only 40 is invalid.
        int rowbase = mt * 16 + ((lane >> 4) << 3);
        if (rowbase < THALO) {
#pragma unroll
            for (int v = 0; v < 8; v++) {
                int tt2 = rowbase + v;                       // <= 39, always in range
                bool tok = (t0 + tt2) < T_LEN;
                float real = d_rr[v] - d_ii[v] + (csbr - csbi);
                float imag = d_ir[v] + d_ri[v] + (csbr + csbi);
                float o_r = tok ? fast_tanh(real) : 0.f;     // branchless select
                float o_i = tok ? fmaxf(imag, 0.f) : 0.f;
                sm_t1[((fi * THALO + tt2) * 2 + 0) * CH_INNER + co] = o_r;
                sm_t1[((fi * THALO + tt2) * 2 + 1) * CH_INNER + co] = o_i;
            }
        }
    }
    __syncthreads();

    // ---- Phase C: depthwise 3-tap complex conv over frequency + cAct (center freq)
    for (int i = tid; i < THALO * CH_INNER; i += 128) {
        int tt = i / CH_INNER;
        int c  = i % CH_INNER;
        int t  = t0 + tt;
        float wrxr = 0.f, wrxi = 0.f, wixr = 0.f, wixi = 0.f;
#pragma unroll
        for (int kf = 0; kf < FREQ_K; kf++) {
            float wr = fcv_wr[c * FREQ_K + kf], wi = fcv_wi[c * FREQ_K + kf];
            float xr = sm_t1[((kf * THALO + tt) * 2 + 0) * CH_INNER + c];
            float xi = sm_t1[((kf * THALO + tt) * 2 + 1) * CH_INNER + c];
            wrxr += wr * xr; wrxi += wr * xi;
            wixr += wi * xr; wixi += wi * xi;
        }
        float bre = fcv_br[c], bie = fcv_bi[c];
        float real = wrxr - wixi + (bre - bie);
        float imag = wrxi + wixr + (bre + bie);
        bool tok = t < T_LEN;                   // zero beyond T -> time-conv zero pad
        sm_t2[(tt * 2 + 0) * CH_INNER + c] = tok ? fast_tanh(real) : 0.f;
        sm_t2[(tt * 2 + 1) * CH_INNER + c] = tok ? fmaxf(imag, 0.f) : 0.f;
    }
    // ---- restage sm_w with tc weights (cs weights dead after barrier above)
    for (int i = tid; i < TC_W_ELEMS; i += 128) {
        int sel = i / (CH_IN * TIME_K * CH_INNER);
        int r   = i % (CH_IN * TIME_K * CH_INNER);
        int co  = r / (TIME_K * CH_INNER);
        int r2  = r % (TIME_K * CH_INNER);
        int tap = r2 / CH_INNER;
        int ci  = r2 % CH_INNER;
        const float* src = sel ? tc_wi : tc_wr;
        sm_w[i] = src[((fc * CH_IN + co) * CH_INNER + ci) * TIME_K + tap];
    }
    __syncthreads();

    // ---- Phase D: time conv as f32 WMMA GEMM (K = 160 = 5 taps x 32 ch).
    // wave = (m-tile, part): A = 16x160 of Xr or Xi, B cols = [Wr co0..7 | Wi co0..7]
    {
        const int mt   = wave >> 1;
        const int p    = wave & 1;
        const int koff = (lane >> 4) << 1;
        const int tl   = mt * 16 + (lane & 15);
        const int n    = lane & 15;
        const int sel  = n >> 3;
        const int co   = n & 7;
        v8f acc = {};
#pragma unroll
        for (int kk = 0; kk < 40; kk++) {
            int kb  = kk * 4 + koff;    // group of 4 never crosses a tap boundary
            int tap = kb >> 5;
            int ci  = kb & 31;
            v2f a, bf;
            const float* ta = &sm_t2[((tl + DIL * tap) * 2 + p) * CH_INNER + ci];
            a.x = ta[0]; a.y = ta[1];
            const float* tb = &sm_w[((sel * CH_IN + co) * TIME_K + tap) * CH_INNER + ci];
            bf.x = tb[0]; bf.y = tb[1];
            acc = WMMA_F32(a, bf, acc);
        }
#pragma unroll
        for (int v = 0; v < 8; v++)
            sm_d[wave][v + ((lane >> 4) << 3)][lane & 15] = acc[v];
    }
    __syncthreads();

    // ---- combine complex parts + cAct + cExp -> sm_t3
    for (int i = tid; i < TT * CH_IN; i += 128) {
        int tl = i / CH_IN;
        int co = i % CH_IN;
        int mt = tl >> 4;
        int tm = tl & 15;
        float d_rr = sm_d[mt * 2 + 0][tm][co];      // Xr.Wr
        float d_rw = sm_d[mt * 2 + 0][tm][co + 8];  // Xr.Wi
        float d_ir = sm_d[mt * 2 + 1][tm][co];      // Xi.Wr
        float d_iw = sm_d[mt * 2 + 1][tm][co + 8];  // Xi.Wi
        float bre = tc_br[fc * CH_IN + co], bie = tc_bi[fc * CH_IN + co];
        float real = d_rr - d_iw + (bre - bie);
        float imag = d_ir + d_rw + (bre + bie);
        float aa = fast_tanh(real);
        float bb = fmaxf(imag, 0.f);
        float mag = __expf(aa);
        sm_t3[(tl * 2 + 0) * CH_IN + co] = mag * __cosf(bb);
        sm_t3[(tl * 2 + 1) * CH_IN + co] = mag * __sinf(bb);
    }
    __syncthreads();

    // ---- Phase E: final per-freq complex 1x1 (8 -> 8) as WMMA
    {
        const int mt   = wave >> 1;
        const int p    = wave & 1;
        const int koff = (lane >> 4) << 1;
        const int tl   = mt * 16 + (lane & 15);
        const int n    = lane & 15;
        const int sel  = n >> 3;
        const int co   = n & 7;
        const float* wsrc = sel ? lw_i : lw_r;
        const int wb = (fc * CH_IN + co) * CH_IN;
        v8f acc = {};
#pragma unroll
        for (int kk = 0; kk < 2; kk++) {
            int ci = kk * 4 + koff;
            v2f a, bf;
            a.x = sm_t3[(tl * 2 + p) * CH_IN + ci];
            a.y = sm_t3[(tl * 2 + p) * CH_IN + ci + 1];
            bf.x = wsrc[wb + ci];
            bf.y = wsrc[wb + ci + 1];
            acc = WMMA_F32(a, bf, acc);
        }
#pragma unroll
        for (int v = 0; v < 8; v++)
            sm_d[wave][v + ((lane >> 4) << 3)][lane & 15] = acc[v];
    }
    __syncthreads();

    // ---- final combine + bias -> global output
    for (int i = tid; i < TT * CH_IN; i += 128) {
        int tl = i / CH_IN;
        int co = i % CH_IN;
        int t  = t0 + tl;
        if (t >= T_LEN) continue;
        int mt = tl >> 4;
        int tm = tl & 15;
        float d_rr = sm_d[mt * 2 + 0][tm][co];
        float d_rw = sm_d[mt * 2 + 0][tm][co + 8];
        float d_ir = sm_d[mt * 2 + 1][tm][co];
        float d_iw = sm_d[mt * 2 + 1][tm][co + 8];
        float bre = lb_r[fc * CH_IN + co], bie = lb_i[fc * CH_IN + co];
        long ob = (((long)(b * 2) * CH_IN + co) * FREQS + fc) * (long)T_LEN + t;
        out[ob]           = d_rr - d_iw + (bre - bie);
        out[ob + PSTRIDE] = d_ir + d_rw + (bre + bie);
    }
}

extern "C" void kernel_launch(void* const* d_in, const int* in_sizes, int n_in,
                              void* d_out, int out_size, void* d_ws, size_t ws_size,
                              hipStream_t stream) {
    (void)in_sizes; (void)n_in; (void)d_ws; (void)ws_size; (void)out_size;
    const float* p[17];
    for (int i = 0; i < 17; i++) p[i] = (const float*)d_in[i];
    dim3 grid(FREQS, (T_LEN + TT - 1) / TT, B_SZ);   // 256 x 32 x 4 blocks
    dim3 block(128);                                  // 4 wave32 waves
    hipLaunchKernelGGL(minibf_fused, grid, block, 0, stream,
                       p[0], p[1], p[2], p[3], p[4], p[5], p[6], p[7], p[8],
                       p[9], p[10], p[11], p[12], p[13], p[14], p[15], p[16],
                       (float*)d_out);
}